// RankFilter2d_37538014167660
// MI455X (gfx1250) — compile-verified
//
#include <hip/hip_runtime.h>
#include <stdint.h>

#ifndef __has_builtin
#define __has_builtin(x) 0
#endif

// Problem constants (fixed by the reference: (1, 2048, 2048) f32, 5x5, rank 13).
#define IMG_H 2048
#define IMG_W 2048
#define TILE  32
#define HALO  2
#define LWID  (TILE + 2 * HALO)   // 36 : logical halo tile width
#define RS    40                  // LDS row stride (dwords): interior 16B-aligned, conflict-free
#define COFF  2                   // halo starts at LDS column 2
#define NTHREADS 256              // 8 wave32s per block

// Pointer types per hipcc diagnostics: b32 takes int*, b64 int2*, b128 int4*,
// global side addrspace(1), LDS side addrspace(3).
typedef int v2i __attribute__((ext_vector_type(2)));
typedef int v4i __attribute__((ext_vector_type(4)));
typedef __attribute__((address_space(1))) int* as1_ip;
typedef __attribute__((address_space(3))) int* as3_ip;
typedef __attribute__((address_space(1))) v2i* as1_v2ip;
typedef __attribute__((address_space(3))) v2i* as3_v2ip;
typedef __attribute__((address_space(1))) v4i* as1_v4ip;
typedef __attribute__((address_space(3))) v4i* as3_v4ip;

// ---- CDNA5 async global->LDS DMA (ASYNCcnt path) -------------------------
__device__ __forceinline__ void async_b32_to_lds(const float* g, float* l) {
#if __has_builtin(__builtin_amdgcn_global_load_async_to_lds_b32)
  __builtin_amdgcn_global_load_async_to_lds_b32(
      (as1_ip)(uintptr_t)g, (as3_ip)(uint32_t)(uintptr_t)l, 0, 0);
#else
  const uint32_t loff = (uint32_t)(uintptr_t)l;
  const uint64_t ga   = (uint64_t)(uintptr_t)g;
  asm volatile("global_load_async_to_lds_b32 %0, %1, off"
               :: "v"(loff), "v"(ga) : "memory");
#endif
}

__device__ __forceinline__ void async_b64_to_lds(const float* g, float* l) {
#if __has_builtin(__builtin_amdgcn_global_load_async_to_lds_b64)
  __builtin_amdgcn_global_load_async_to_lds_b64(
      (as1_v2ip)(uintptr_t)g, (as3_v2ip)(uint32_t)(uintptr_t)l, 0, 0);
#else
  const uint32_t loff = (uint32_t)(uintptr_t)l;
  const uint64_t ga   = (uint64_t)(uintptr_t)g;
  asm volatile("global_load_async_to_lds_b64 %0, %1, off"
               :: "v"(loff), "v"(ga) : "memory");
#endif
}

__device__ __forceinline__ void async_b128_to_lds(const float* g, float* l) {
#if __has_builtin(__builtin_amdgcn_global_load_async_to_lds_b128)
  __builtin_amdgcn_global_load_async_to_lds_b128(
      (as1_v4ip)(uintptr_t)g, (as3_v4ip)(uint32_t)(uintptr_t)l, 0, 0);
#else
  const uint32_t loff = (uint32_t)(uintptr_t)l;
  const uint64_t ga   = (uint64_t)(uintptr_t)g;
  asm volatile("global_load_async_to_lds_b128 %0, %1, off"
               :: "v"(loff), "v"(ga) : "memory");
#endif
}

__device__ __forceinline__ void wait_async_zero() {
#if __has_builtin(__builtin_amdgcn_s_wait_asynccnt)
  __builtin_amdgcn_s_wait_asynccnt(0);
#else
  asm volatile("s_wait_asynccnt 0" ::: "memory");
#endif
}

// Compare-exchange on distinct 64-bit keys: v[a]=min, v[b]=max.
#define S2(a, b)                                                   \
  {                                                                \
    const uint64_t ta = v[a], tb = v[b];                           \
    const bool c = tb < ta;                                        \
    v[a] = c ? tb : ta;                                            \
    v[b] = c ? ta : tb;                                            \
  }

__global__ __launch_bounds__(NTHREADS)
void rank13_of_25_kernel(const float* __restrict__ x,
                         float* __restrict__ out_val,
                         int*   __restrict__ out_idx) {
  __shared__ float tile[LWID * RS];   // 36 rows x 40 dwords = 5760 B

  const int tx0 = blockIdx.x * TILE;
  const int ty0 = blockIdx.y * TILE;
  const int tid = threadIdx.x;

  // ---- Stage 36x36 halo tile into LDS (async DMA, no VGPR staging). ------
  const bool interior = (tx0 >= HALO) && (tx0 + TILE + HALO <= IMG_W) &&
                        (ty0 >= HALO) && (ty0 + TILE + HALO <= IMG_H);
  if (interior) {
    // Wide path (94% of blocks): per row = b64 | 8 x b128 | b64, all in-bounds.
    const float* gbase = x + (size_t)(ty0 - HALO) * IMG_W + tx0;
    // 288 aligned 16B chunks: chunk c -> (row = c>>3, quad = c&7).
    for (int c = tid; c < 8 * LWID; c += NTHREADS) {
      const int row = c >> 3, quad = c & 7;
      async_b128_to_lds(gbase + (size_t)row * IMG_W + quad * 4,
                        &tile[row * RS + COFF + HALO + quad * 4]);  // byte 160r+16+16q
    }
    // 72 halo b64 chunks: tid<72 -> (row = tid>>1, side = tid&1).
    if (tid < 2 * LWID) {
      const int row = tid >> 1, side = tid & 1;
      const int gxo = side ? TILE : -HALO;          // right / left halo pair
      const int lco = side ? (COFF + HALO + TILE) : COFF;
      async_b64_to_lds(gbase + (size_t)row * IMG_W + gxo, &tile[row * RS + lco]);
    }
  } else {
    // Edge path: scalar b32 with bounds check; OOB -> 0.0f (constant pad).
    for (int k = tid; k < LWID * LWID; k += NTHREADS) {
      const int ly = k / LWID;
      const int lx = k - ly * LWID;
      const int gy = ty0 - HALO + ly;
      const int gx = tx0 - HALO + lx;
      float* dst = &tile[ly * RS + COFF + lx];
      if ((unsigned)gy < (unsigned)IMG_H && (unsigned)gx < (unsigned)IMG_W) {
        async_b32_to_lds(x + (size_t)gy * IMG_W + gx, dst);
      } else {
        *dst = 0.0f;                  // ds_store; disjoint slots from the async path
      }
    }
  }
  wait_async_zero();                  // my wave's DMAs have landed in LDS
  __syncthreads();                    // (also waits DScnt) -> whole tile visible

  // ---- Each thread computes 4 pixels (32x32 / 256). ----------------------
  const int cx  = tid & (TILE - 1);   // 0..31
  const int cy0 = tid >> 5;           // 0..7
#pragma unroll 1                      // keep one 50-VGPR network live; 8 waves hide latency
  for (int r = 0; r < 4; ++r) {
    const int cy = cy0 + 8 * r;

    // Build 25 distinct 64-bit keys: (order-preserving u32 of f32) << 5 | window_idx.
    // Distinct keys => unique rank-12 element == JAX stable-argsort 13th element,
    // so value AND stable kth index come out of one selection network.
    uint64_t v[25];
#pragma unroll
    for (int di = 0; di < 5; ++di) {
#pragma unroll
      for (int dj = 0; dj < 5; ++dj) {
        const int w = di * 5 + dj;
        const uint32_t u = __float_as_uint(tile[(cy + di) * RS + COFF + cx + dj]);
        const uint32_t s = u ^ ((uint32_t)((int32_t)u >> 31) | 0x80000000u);
        v[w] = ((uint64_t)s << 5) | (uint32_t)w;
      }
    }

    // McGuire (ShaderX6 2006) median-of-25 selection network: 70 exchanges.
    S2(0, 1)   S2(3, 4)   S2(2, 4)   S2(2, 3)   S2(6, 7)
    S2(5, 7)   S2(5, 6)   S2(9, 7)   S2(1, 7)   S2(1, 4)
    S2(12, 13) S2(11, 13) S2(11, 12) S2(15, 16) S2(14, 16)
    S2(14, 15) S2(18, 19) S2(17, 19) S2(17, 18) S2(21, 22)
    S2(20, 22) S2(20, 21) S2(23, 24) S2(2, 5)   S2(3, 6)
    S2(0, 6)   S2(0, 3)   S2(4, 7)   S2(1, 7)   S2(1, 4)
    S2(11, 14) S2(8, 14)  S2(8, 11)  S2(12, 15) S2(9, 15)
    S2(9, 12)  S2(13, 16) S2(10, 16) S2(10, 13) S2(20, 23)
    S2(17, 23) S2(17, 20) S2(21, 24) S2(18, 24) S2(18, 21)
    S2(19, 22) S2(8, 17)  S2(9, 18)  S2(0, 18)  S2(0, 9)
    S2(10, 19) S2(1, 19)  S2(1, 10)  S2(11, 20) S2(2, 20)
    S2(2, 11)  S2(12, 21) S2(3, 21)  S2(3, 12)  S2(13, 22)
    S2(4, 22)  S2(4, 13)  S2(14, 23) S2(5, 23)  S2(5, 14)
    S2(15, 24) S2(6, 24)  S2(6, 15)  S2(7, 16)  S2(7, 8)

    const uint64_t m = v[12];                 // rank-12 (0-indexed) key
    const uint32_t sk = (uint32_t)(m >> 5);
    const uint32_t bits = (sk & 0x80000000u) ? (sk ^ 0x80000000u) : ~sk;  // undo sortable map
    const size_t p = (size_t)(ty0 + cy) * IMG_W + (tx0 + cx);
    out_val[p] = __uint_as_float(bits);
    out_idx[p] = (int)(m & 31u);
  }
}

extern "C" void kernel_launch(void* const* d_in, const int* in_sizes, int n_in,
                              void* d_out, int out_size, void* d_ws, size_t ws_size,
                              hipStream_t stream) {
  (void)in_sizes; (void)n_in; (void)out_size; (void)d_ws; (void)ws_size;
  const float* x = (const float*)d_in[0];
  // Tuple output, concatenated flat: values (H*W f32) then kth_idx (H*W i32).
  float* out_val = (float*)d_out;
  int*   out_idx = (int*)d_out + (size_t)IMG_H * IMG_W;

  dim3 grid(IMG_W / TILE, IMG_H / TILE);   // 64 x 64 tiles
  rank13_of_25_kernel<<<grid, NTHREADS, 0, stream>>>(x, out_val, out_idx);
}